// BertEmbedding_60327110640046
// MI455X (gfx1250) — compile-verified
//
#include <hip/hip_runtime.h>
#include <stdint.h>

// -------- types --------
typedef __attribute__((ext_vector_type(4))) float        f32x4;
typedef __attribute__((ext_vector_type(4))) unsigned int u32x4;
typedef __attribute__((ext_vector_type(8))) int          i32x8;
typedef __attribute__((ext_vector_type(4))) int          i32x4;

#define HDIM 1024   // hidden size
#define SEQ  512    // sequence length (power of two)
#define EPSV 1e-5f

// One workgroup (256 threads = 8 wave32) per (b,s) row.
// Wave 0 issues a TDM tensor_load_to_lds for the 4KB token embedding row
// (workgroup-uniform gather), waits TENSORcnt, then the whole block does the
// add + wave32-shuffle LayerNorm and streams the result out with NT b128 stores.
__global__ __launch_bounds__(256)
void bert_embed_ln_kernel(const int* __restrict__ input_ids,
                          const int* __restrict__ token_type_ids,
                          const float* __restrict__ token_W,
                          const float* __restrict__ pos_W,
                          const float* __restrict__ type_W,
                          const float* __restrict__ ln_w,
                          const float* __restrict__ ln_b,
                          float* __restrict__ out)
{
    __shared__ __attribute__((aligned(16))) float s_tok[HDIM]; // 4KB staged token row
    __shared__ float s_red[16];                                // 8 wave sums + 8 wave sumsqs

    const int row  = blockIdx.x;         // b*S + s
    const int spos = row & (SEQ - 1);    // s
    const int tid  = threadIdx.x;
    const int lane = tid & 31;
    const int wave = tid >> 5;

    // Uniform scalar loads (one token id / type id per row)
    const int tok_id  = input_ids[row];
    const int type_id = token_type_ids[row];

    if (wave == 0) {
        // ---- Build Tensor DMA Descriptor (D#) for 1 x 1024 f32 row ----
        const uint64_t ga  = (uint64_t)(uintptr_t)(token_W + (size_t)tok_id * HDIM);
        const uint32_t lds = (uint32_t)(uintptr_t)(void*)s_tok; // low 32 bits = WG-relative LDS offset

        // Group 0: count=1 | lds_addr | global_addr[56:0] | type=2
        u32x4 g0;
        g0[0] = 1u;                                               // count = 1 (valid), user mode
        g0[1] = lds;                                              // lds_addr (bytes)
        g0[2] = (uint32_t)ga;                                     // global_addr[31:0]
        g0[3] = ((uint32_t)(ga >> 32) & 0x01FFFFFFu) | (2u << 30); // global_addr[56:32] | type=2

        // Group 1: data_size=4B, tensor_dim0=1024, tensor_dim1=1,
        //          tile_dim0=1024, tile_dim1=1, dim0_stride=1024
        i32x8 g1;
        g1[0] = 0x00020000;                   // wg_mask=0, data_size=2 (4 bytes)
        g1[1] = (int)((uint32_t)HDIM << 16);  // [47:32]=0, [63:48]=tensor_dim0 lo16 (1024)
        g1[2] = (int)(1u << 16);              // [79:64]=tensor_dim0 hi16=0, [95:80]=tensor_dim1 lo16=1
        g1[3] = (int)((uint32_t)HDIM << 16);  // [111:96]=tensor_dim1 hi16=0, [127:112]=tile_dim0=1024
        g1[4] = 1;                            // [143:128]=tile_dim1=1, [159:144]=tile_dim2=0
        g1[5] = HDIM;                         // tensor_dim0_stride[31:0] = 1024
        g1[6] = 0;                            // stride0 hi16 = 0, tensor_dim1_stride lo16 = 0
        g1[7] = 0;                            // tensor_dim1_stride hi = 0

        const i32x4 gz4 = {0, 0, 0, 0};            // groups 2/3 unused (<=2D tensor)
        const i32x8 gz8 = {0, 0, 0, 0, 0, 0, 0, 0}; // trailing group (clang-23 6-arg form)

        __builtin_amdgcn_tensor_load_to_lds(g0, g1, gz4, gz4, gz8, /*cpol=*/0);
        __builtin_amdgcn_s_wait_tensorcnt(0);
    }
    __syncthreads();   // publish TDM-written LDS to all 8 waves

    // ---- Gather-add: e = token + type + position (b128 per thread) ----
    const int col = tid * 4;
    f32x4 e = *(const f32x4*)(s_tok + col);
    e = e + *(const f32x4*)(type_W + (size_t)type_id * HDIM + col);
    e = e + *(const f32x4*)(pos_W  + (size_t)spos    * HDIM + col);

    // ---- LayerNorm reduction: wave32 shfl_xor tree, then 8-wave combine ----
    float sum = e[0] + e[1] + e[2] + e[3];
    float ssq = e[0]*e[0] + e[1]*e[1] + e[2]*e[2] + e[3]*e[3];
#pragma unroll
    for (int off = 16; off > 0; off >>= 1) {
        sum += __shfl_xor(sum, off, 32);
        ssq += __shfl_xor(ssq, off, 32);
    }
    if (lane == 0) { s_red[wave] = sum; s_red[8 + wave] = ssq; }
    __syncthreads();

    float tsum = 0.0f, tssq = 0.0f;
#pragma unroll
    for (int w = 0; w < 8; ++w) { tsum += s_red[w]; tssq += s_red[8 + w]; }

    const float mean = tsum * (1.0f / HDIM);
    const float var  = tssq * (1.0f / HDIM) - mean * mean;
    const float rstd = rsqrtf(var + EPSV);

    // ---- Affine + streaming NT store ----
    const f32x4 w4 = *(const f32x4*)(ln_w + col);
    const f32x4 b4 = *(const f32x4*)(ln_b + col);
    f32x4 o = (e - mean) * rstd * w4 + b4;

    __builtin_nontemporal_store(o, (f32x4*)(out + (size_t)row * HDIM + col));
}

extern "C" void kernel_launch(void* const* d_in, const int* in_sizes, int n_in,
                              void* d_out, int out_size, void* d_ws, size_t ws_size,
                              hipStream_t stream) {
    (void)n_in; (void)out_size; (void)d_ws; (void)ws_size;
    const int*   input_ids      = (const int*)  d_in[0];
    const int*   token_type_ids = (const int*)  d_in[1];
    const float* token_W        = (const float*)d_in[2];
    const float* pos_W          = (const float*)d_in[3];
    const float* type_W         = (const float*)d_in[4];
    const float* ln_w           = (const float*)d_in[5];
    const float* ln_b           = (const float*)d_in[6];
    float*       out            = (float*)d_out;

    const int rows = in_sizes[0];          // B*S = 16384
    dim3 grid(rows), block(256);           // 8 wave32 per row, 4 floats/thread
    hipLaunchKernelGGL(bert_embed_ln_kernel, grid, block, 0, stream,
                       input_ids, token_type_ids, token_W, pos_W, type_W,
                       ln_w, ln_b, out);
}